// LTFWG_10977936409018
// MI455X (gfx1250) — compile-verified
//
#include <hip/hip_runtime.h>

typedef __attribute__((ext_vector_type(2))) float v2f;
typedef __attribute__((ext_vector_type(8))) float v8f;

#define KDEG   16
#define NFEAT  128
#define NTEMPL 16
#define LDSTR  132   // 128 + 4 pad -> row m starts at bank 4m (no 16-way conflict)

// ws float layout: [0,2048) tmean | [2048,2064) tsq | [2064,2080) mCt | [2080,2096) mCt2

__global__ void ltfwg_tstats(const float* __restrict__ tmpl,
                             const float* __restrict__ tfeat,
                             float* __restrict__ ws) {
    int u = blockIdx.x * blockDim.x + threadIdx.x;
    float* tmean = ws;
    float* tsq   = ws + 2048;
    float* mCt   = ws + 2064;
    float* mCt2  = ws + 2080;
    if (u < NTEMPL * NFEAT) {                 // tmean[t][j] = mean over 8 template nodes
        int t = u >> 7, j = u & 127;
        float s = 0.f;
        #pragma unroll
        for (int c = 0; c < 8; ++c) s += tfeat[(t * 8 + c) * NFEAT + j];
        tmean[t * NFEAT + j] = s * 0.125f;
    }
    if (u < NTEMPL) {                         // tsq, mCt, mCt2 (tiny, serial per template)
        float s2 = 0.f;
        for (int c = 0; c < 8; ++c)
            for (int j = 0; j < NFEAT; ++j) {
                float v = tfeat[(u * 8 + c) * NFEAT + j];
                s2 += v * v;
            }
        tsq[u] = s2 * 0.125f;
        float s = 0.f, q = 0.f;
        for (int e = 0; e < 64; ++e) {
            float v = tmpl[u * 64 + e];
            s += v; q += v * v;
        }
        mCt[u]  = s * (1.f / 64.f);
        mCt2[u] = q * (1.f / 64.f);
    }
}

__global__ __launch_bounds__(256)
void ltfwg_main(const float* __restrict__ x,
                const int*   __restrict__ dst,   // edge_index row 1, [N*K]
                const float* __restrict__ ws,
                float*       __restrict__ out,
                int nnodes) {
    __shared__ float lds_f[16 * LDSTR];  // fmean tile [16][128], padded stride
    __shared__ float lds_sq[16];
    __shared__ float lds_mc[16];

    const int tid  = threadIdx.x;
    const int lane = tid & 31;
    const int wave = tid >> 5;
    const int tile = blockIdx.x;

    for (int i = tid; i < 16 * LDSTR; i += 256) lds_f[i] = 0.f;
    if (tid < 16) { lds_sq[tid] = 0.f; lds_mc[tid] = 0.f; }
    __syncthreads();

    // ---------- Phase 1: per-node gather + reductions (one wave per 2 nodes) ----------
    for (int s = 0; s < 2; ++s) {
        const int ln = wave * 2 + s;          // local node 0..15
        const int n  = tile * 16 + ln;
        if (n < nnodes) {
            // all 32 lanes hold nbr[lane&15] so __shfl(.., a) broadcasts nbr[a]
            const int my_nbr = dst[n * KDEG + (lane & 15)];

            // feature gather: lane owns columns 4*lane..4*lane+3 (float4 per row)
            float4 fsum = make_float4(0.f, 0.f, 0.f, 0.f);
            float sqacc = 0.f;
            #pragma unroll
            for (int a = 0; a < KDEG; ++a) {
                const int na = __shfl(my_nbr, a, 32);
                const float4 v = *(const float4*)(x + (size_t)na * NFEAT + lane * 4);
                fsum.x += v.x; fsum.y += v.y; fsum.z += v.z; fsum.w += v.w;
                sqacc += v.x * v.x + v.y * v.y + v.z * v.z + v.w * v.w;
            }
            #pragma unroll
            for (int m = 16; m >= 1; m >>= 1) sqacc += __shfl_xor(sqacc, m, 32);
            if (lane == 0) lds_sq[ln] = sqacc * (1.f / 16.f);

            float4 fm;
            fm.x = fsum.x * (1.f / 16.f); fm.y = fsum.y * (1.f / 16.f);
            fm.z = fsum.z * (1.f / 16.f); fm.w = fsum.w * (1.f / 16.f);
            *(float4*)(&lds_f[ln * LDSTR + lane * 4]) = fm;

            // structure: lane covers a = lane&15, b in [(lane>>4)*8, +8) -> 256 pairs
            const int4* nrow = (const int4*)(dst + (size_t)my_nbr * KDEG);
            const int4 r0 = nrow[0], r1 = nrow[1], r2 = nrow[2], r3 = nrow[3];
            int cnt = 0;
            #pragma unroll
            for (int bi = 0; bi < 8; ++bi) {
                const int b  = ((lane >> 4) << 3) + bi;
                const int nb = __shfl(my_nbr, b, 32);
                const bool hit =
                    (r0.x == nb) | (r0.y == nb) | (r0.z == nb) | (r0.w == nb) |
                    (r1.x == nb) | (r1.y == nb) | (r1.z == nb) | (r1.w == nb) |
                    (r2.x == nb) | (r2.y == nb) | (r2.z == nb) | (r2.w == nb) |
                    (r3.x == nb) | (r3.y == nb) | (r3.z == nb) | (r3.w == nb);
                cnt += hit ? 1 : 0;
            }
            #pragma unroll
            for (int m = 16; m >= 1; m >>= 1) cnt += __shfl_xor(cnt, m, 32);
            if (lane == 0) lds_mc[ln] = (float)cnt * (1.f / 256.f);
        }
    }
    __syncthreads();

    // ---------- Phase 2: wave 0 computes 16x16 tile via f32 WMMA over K=128 ----------
    if (wave == 0) {
        const float* tmean = ws;
        const float* tsq   = ws + 2048;
        const float* mCt   = ws + 2064;
        const float* mCt2  = ws + 2080;

        const int mrow = lane & 15;           // A: M-row ; B: N-col (template id)
        const int kb   = (lane >> 4) * 2;     // K sub-offset per half-wave

        v8f c = {0.f, 0.f, 0.f, 0.f, 0.f, 0.f, 0.f, 0.f};
        #pragma unroll
        for (int kk = 0; kk < 32; ++kk) {
            const int k0 = kk * 4 + kb;
            v2f af, bf;
            af.x = lds_f[mrow * LDSTR + k0];
            af.y = lds_f[mrow * LDSTR + k0 + 1];
            bf.x = tmean[mrow * NFEAT + k0];      // B = tmean^T : B[k][t] = tmean[t][k]
            bf.y = tmean[mrow * NFEAT + k0 + 1];
            c = __builtin_amdgcn_wmma_f32_16x16x4_f32(
                    false, af, false, bf, (short)0, c, false, false);
        }

        const int t = lane & 15;
        const float tq = tsq[t], mt = mCt[t], mt2 = mCt2[t];
        const int mhi = (lane >> 4) << 3;
        #pragma unroll
        for (int r = 0; r < 8; ++r) {
            const int m = r + mhi;
            const int n = tile * 16 + m;
            if (n < nnodes) {
                const float dot  = c[r];                       // fmean . tmean[t]
                const float feat = lds_sq[m] + tq - 2.f * dot;
                const float mc   = lds_mc[m];
                const float st   = mc + mt2 - 2.f * mc * mt;
                out[n * NTEMPL + t] = 0.5f * feat + 0.5f * st; // ALPHA = 0.5
            }
        }
    }
}

extern "C" void kernel_launch(void* const* d_in, const int* in_sizes, int n_in,
                              void* d_out, int out_size, void* d_ws, size_t ws_size,
                              hipStream_t stream) {
    const float* x     = (const float*)d_in[0];
    const int*   edge  = (const int*)  d_in[1];
    const float* tmpl  = (const float*)d_in[2];
    const float* tfeat = (const float*)d_in[3];
    float* out = (float*)d_out;
    float* ws  = (float*)d_ws;

    const int nnodes = in_sizes[0] / NFEAT;
    const int* dstrow = edge + in_sizes[1] / 2;   // edge_index[1]

    ltfwg_tstats<<<8, 256, 0, stream>>>(tmpl, tfeat, ws);

    const int tiles = (nnodes + 15) / 16;         // 3125 for N=50000 (exact)
    ltfwg_main<<<tiles, 256, 0, stream>>>(x, dstrow, ws, out, nnodes);
}